// LovaszSoftmax_9543417332109
// MI455X (gfx1250) — compile-verified
//
#include <hip/hip_runtime.h>
#include <hip/hip_bf16.h>

// Lovasz-Softmax via exact tie-telescoping + 2^16-bin counting sort.
// B=8, C=21, H=W=512.  Segments = B*C = 168, pixels per segment = 262144.

#define CCLS 21
#define NIMG 8
#define PIX  262144           // 512*512 = 2^18
#define NSEG (NIMG * CCLS)    // 168
#define BINS 65536

typedef __attribute__((ext_vector_type(2))) float v2f;
typedef __attribute__((ext_vector_type(8))) float v8f;

// ---------------------------------------------------------------- zero ws
__global__ void k_zero(uint4* __restrict__ p, unsigned n4) {
    unsigned i = blockIdx.x * blockDim.x + threadIdx.x;
    unsigned stride = gridDim.x * blockDim.x;
    uint4 z = {0u, 0u, 0u, 0u};
    for (; i < n4; i += stride) p[i] = z;
}

// ------------------------------------------------- softmax + histogram
__global__ __launch_bounds__(256) void k_hist(const float* __restrict__ logits,
                                              const int* __restrict__ labels,
                                              unsigned* __restrict__ histTot,
                                              unsigned* __restrict__ histFg) {
    int gid = blockIdx.x * 256 + threadIdx.x;   // exactly NIMG*PIX threads
    int b = gid >> 18;                          // / PIX
    int p = gid & (PIX - 1);

    const float* base = logits + (size_t)b * CCLS * PIX + p;
    float v[CCLS];
    float m = -3.4e38f;
#pragma unroll
    for (int c = 0; c < CCLS; ++c) {
        float x = __builtin_nontemporal_load(base + (size_t)c * PIX);
        v[c] = x;
        m = fmaxf(m, x);
    }
    float s = 0.0f;
#pragma unroll
    for (int c = 0; c < CCLS; ++c) {
        v[c] = __expf(v[c] - m);
        s += v[c];
    }
    float inv = 1.0f / s;
    int lab = __builtin_nontemporal_load(labels + gid);

#pragma unroll
    for (int c = 0; c < CCLS; ++c) {
        float prob = v[c] * inv;
        bool fg = (c == lab);
        float e = fg ? (1.0f - prob) : prob;
        int bin = (int)(e * 65536.0f);
        bin = bin < 0 ? 0 : (bin > 65535 ? 65535 : bin);
        unsigned idx = (unsigned)(b * CCLS + c) * (unsigned)BINS + (unsigned)bin;
        atomicAdd(&histTot[idx], 1u);
        if (fg) atomicAdd(&histFg[idx], 1u);
    }
}

// -------- 16 row-sums of a 256-float LDS array via V_WMMA_F32_16X16X4_F32.
// A = X (16x16 as 4 chained K=4 blocks, documented layout), B = all-ones
// (layout-invariant), C/D layout documented: VGPR v / lane l -> M = v+8*(l>=16).
// Executed redundantly by ALL waves (full EXEC); identical LDS writes benign.
__device__ __forceinline__ void rowsum16(const float* __restrict__ lds256,
                                         float* __restrict__ out16) {
#if defined(__gfx1250__) && __has_builtin(__builtin_amdgcn_wmma_f32_16x16x4_f32)
    int lane = threadIdx.x & 31;
    int m = lane & 15;
    int hi = lane >> 4;              // 0: lanes 0-15 (K=0,1) ; 1: lanes 16-31 (K=2,3)
    v8f acc = {};
    v2f bones;
    bones.x = 1.0f;
    bones.y = 1.0f;
#pragma unroll
    for (int kb = 0; kb < 4; ++kb) {
        int k0 = 4 * kb + (hi ? 2 : 0);
        v2f a;
        a.x = lds256[16 * m + k0];
        a.y = lds256[16 * m + k0 + 1];
        acc = __builtin_amdgcn_wmma_f32_16x16x4_f32(false, a, false, bones,
                                                    (short)0, acc, false, false);
    }
    if (m == 0) {                    // lane 0 -> rows 0..7, lane 16 -> rows 8..15
#pragma unroll
        for (int v = 0; v < 8; ++v) out16[v + 8 * hi] = acc[v];
    }
#else
    int lane = threadIdx.x & 31;
    if (lane < 16) {
        float s = 0.0f;
#pragma unroll
        for (int c = 0; c < 16; ++c) s += lds256[16 * lane + c];
        out16[lane] = s;
    }
#endif
}

// ------------------------------------------ per-segment Lovasz traversal
__global__ __launch_bounds__(256) void k_lovasz(const unsigned* __restrict__ histTot,
                                                const unsigned* __restrict__ histFg,
                                                float* __restrict__ segLoss) {
    __shared__ float tN[256], tF[256], partial[256];
    __shared__ float rowN[16], rowF[16], offN[16], offF[16], rowP[16];
    __shared__ float Gsh;

    int t = threadIdx.x;
    int seg = blockIdx.x;
    const unsigned* HT = histTot + (size_t)seg * BINS;
    const unsigned* HF = histFg + (size_t)seg * BINS;

    // Phase A: chunk totals. Descending-e order: q = 256*t + j, bin = 65535 - q.
    int binBase = 65535 - 256 * t;
    float nS = 0.0f, fS = 0.0f;
    for (int j = 0; j < 256; ++j) {
        int bin = binBase - j;
        nS += (float)HT[bin];       // counts <= 2^18: exact in f32
        fS += (float)HF[bin];
    }
    tN[t] = nS;
    tF[t] = fS;
    __syncthreads();

    // Phase B: 256 -> 16 row-sums via WMMA, then scalar exclusive scan of 16.
    rowsum16(tN, rowN);
    rowsum16(tF, rowF);
    __syncthreads();
    if (t == 0) {
        float aN = 0.0f, aF = 0.0f;
        for (int r = 0; r < 16; ++r) {
            offN[r] = aN; aN += rowN[r];
            offF[r] = aF; aF += rowF[r];
        }
        Gsh = aF;                    // G = total foreground count of this class
    }
    __syncthreads();

    float G = Gsh;
    int r = t >> 4;
    float pos = offN[r], F = offF[r];
    for (int c = 16 * r; c < t; ++c) {  // intra-row exclusive prefix (<=15 adds)
        pos += tN[c];
        F += tF[c];
    }

    // Phase C: traverse this thread's 256 bins; telescoped per-bin contribution.
    float jPrev = (pos <= 0.0f) ? 0.0f : (1.0f - (G - F) / (G + pos - F));
    float lsum = 0.0f;
    for (int j = 0; j < 256; ++j) {
        int bin = binBase - j;
        float n = (float)HT[bin];
        if (n != 0.0f) {
            float f = (float)HF[bin];
            pos += n;
            F += f;
            float jNew = 1.0f - (G - F) / (G + pos - F);  // denom >= pos >= 1
            float e = ((float)bin + 0.5f) * (1.0f / 65536.0f);
            lsum += e * (jNew - jPrev);
            jPrev = jNew;
        }
    }
    partial[t] = lsum;
    __syncthreads();

    // Phase D: deterministic reduction 256 -> 16 (WMMA) -> 1 (scalar).
    rowsum16(partial, rowP);
    __syncthreads();
    if (t == 0) {
        float s = 0.0f;
        for (int rr = 0; rr < 16; ++rr) s += rowP[rr];
        segLoss[seg] = s;
    }
}

// --------------------------------------------------------- final mean
__global__ void k_final(const float* __restrict__ segLoss, float* __restrict__ out) {
    if (threadIdx.x == 0 && blockIdx.x == 0) {
        float s = 0.0f;
        for (int i = 0; i < NSEG; ++i) s += segLoss[i];
        out[0] = s * (1.0f / (float)NSEG);  // mean over C then B == mean over all 168
    }
}

extern "C" void kernel_launch(void* const* d_in, const int* in_sizes, int n_in,
                              void* d_out, int out_size, void* d_ws, size_t ws_size,
                              hipStream_t stream) {
    const float* logits = (const float*)d_in[0];
    const int* labels = (const int*)d_in[1];

    unsigned* histTot = (unsigned*)d_ws;                        // 168*65536 u32 = 44 MB
    unsigned* histFg = histTot + (size_t)NSEG * BINS;           // +44 MB
    float* segLoss = (float*)(histFg + (size_t)NSEG * BINS);    // +672 B
    float* out = (float*)d_out;

    unsigned histU32 = 2u * NSEG * BINS;                        // 22,020,096 (div by 4)
    k_zero<<<2048, 256, 0, stream>>>((uint4*)histTot, histU32 / 4u);
    k_hist<<<(NIMG * PIX) / 256, 256, 0, stream>>>(logits, labels, histTot, histFg);
    k_lovasz<<<NSEG, 256, 0, stream>>>(histTot, histFg, segLoss);
    k_final<<<1, 32, 0, stream>>>(segLoss, out);
}